// DynamicFiltering_8607114461271
// MI455X (gfx1250) — compile-verified
//
#include <hip/hip_runtime.h>
#include <cstdint>

// Dynamic 3x3 filtering, B=8 C=3 H=W=512, fp32.
// Bandwidth-bound (~276 MB of mandatory traffic, ~0.39 FLOP/byte), so the
// kernel is organized around data movement on CDNA5:
//  - async global->LDS staging of the image strip (ASYNCcnt path)
//  - ds_load reuse of the 3x3 halo from LDS
//  - non-temporal hints for the 226 MB kernels stream and write-once output
//    so the L2 (192 MB) stays dedicated to the 25 MB image.

typedef float v4f __attribute__((ext_vector_type(4)));
typedef int   v4i __attribute__((ext_vector_type(4)));

constexpr int KB = 8;      // batch
constexpr int KC = 3;      // channels
constexpr int KH = 512;
constexpr int KW = 512;
constexpr int KTAPS = 9;   // 3x3
constexpr int TY = 8;                  // output rows per block
constexpr int ROWS = TY + 2;           // staged rows (with vertical halo)
constexpr int LDS_STRIDE = KW + 8;     // floats; row base 16B aligned
constexpr int W4 = KW / 4;             // 128 float4 per row

__device__ __forceinline__ void async_copy_b128(const float* gsrc, float* ldst) {
#if defined(__gfx1250__) && __has_builtin(__builtin_amdgcn_global_load_async_to_lds_b128)
  __builtin_amdgcn_global_load_async_to_lds_b128(
      (__attribute__((address_space(1))) v4i*)gsrc,
      (__attribute__((address_space(3))) v4i*)ldst,
      /*offset=*/0, /*cpol=*/0);
#else
  *(v4f*)ldst = *(const v4f*)gsrc;   // fallback: load + ds_store
#endif
}

__device__ __forceinline__ void wait_async0() {
#if defined(__gfx1250__) && __has_builtin(__builtin_amdgcn_s_wait_asynccnt)
  __builtin_amdgcn_s_wait_asynccnt(0);
#elif defined(__AMDGCN__)
  asm volatile("s_wait_asynccnt 0" ::: "memory");
#endif
}

__global__ __launch_bounds__(256) void dynfilt3x3_kernel(
    const float* __restrict__ img,
    const float* __restrict__ ker,
    float* __restrict__ out) {
  __shared__ float lds[ROWS * LDS_STRIDE];

  const int tid = threadIdx.x;
  constexpr int blocksPerPlane = KH / TY;          // 64
  const int plane = blockIdx.x / blocksPerPlane;   // b*C + c
  const int y0 = (blockIdx.x % blocksPerPlane) * TY;

  const size_t HW = (size_t)KH * KW;
  const float* imgP = img + (size_t)plane * HW;
  const float* kerP = ker + (size_t)plane * KTAPS * HW;
  float* outP = out + (size_t)plane * HW;

  // ---- Stage ROWS padded image rows into LDS (async b128 per lane) ----
  // LDS row r holds global row (y0 + r - 1); image data starts at float
  // index 4 (16B aligned); index 3 is the x=-1 zero pad, index 4+KW the
  // x=KW zero pad.
  for (int i = tid; i < ROWS * W4; i += 256) {
    const int r  = i >> 7;         // 0..9
    const int c4 = i & (W4 - 1);   // 0..127
    const int gy = y0 + r - 1;
    float* dst = &lds[r * LDS_STRIDE + 4 + c4 * 4];
    if (gy >= 0 && gy < KH) {
      async_copy_b128(imgP + (size_t)gy * KW + c4 * 4, dst);
    } else {
      v4f z = {0.f, 0.f, 0.f, 0.f};
      *(v4f*)dst = z;
    }
  }
  if (tid < ROWS) {
    lds[tid * LDS_STRIDE + 3] = 0.f;        // left pad (x = -1)
    lds[tid * LDS_STRIDE + 4 + KW] = 0.f;   // right pad (x = KW)
  }
  wait_async0();
  __syncthreads();

  // ---- Compute: TY*W4 = 1024 float4 tasks, 4 per thread, no divergence ----
  for (int t = tid; t < TY * W4; t += 256) {
    const int r  = t >> 7;         // output row within strip
    const int c4 = t & (W4 - 1);
    const int x  = c4 * 4;
    const int gy = y0 + r;

    const float* kp = kerP + (size_t)gy * KW + x;
    v4f acc = {0.f, 0.f, 0.f, 0.f};

#pragma unroll
    for (int ki = 0; ki < 3; ++ki) {
      // 6-float window covers kj=0..2 for 4 consecutive outputs.
      const int base = (r + ki) * LDS_STRIDE + 3 + x;
      float w0 = lds[base + 0];
      float w1 = lds[base + 1];
      float w2 = lds[base + 2];
      float w3 = lds[base + 3];
      float w4 = lds[base + 4];
      float w5 = lds[base + 5];
      float wv[6] = {w0, w1, w2, w3, w4, w5};
#pragma unroll
      for (int kj = 0; kj < 3; ++kj) {
        const v4f kv = __builtin_nontemporal_load(
            (const v4f*)(kp + (size_t)(ki * 3 + kj) * HW));
        acc[0] += kv[0] * wv[kj + 0];
        acc[1] += kv[1] * wv[kj + 1];
        acc[2] += kv[2] * wv[kj + 2];
        acc[3] += kv[3] * wv[kj + 3];
      }
    }
    __builtin_nontemporal_store(acc, (v4f*)(outP + (size_t)gy * KW + x));
  }
}

extern "C" void kernel_launch(void* const* d_in, const int* in_sizes, int n_in,
                              void* d_out, int out_size, void* d_ws, size_t ws_size,
                              hipStream_t stream) {
  (void)in_sizes; (void)n_in; (void)out_size; (void)d_ws; (void)ws_size;
  const float* img = (const float*)d_in[0];   // [8,3,512,512]
  const float* ker = (const float*)d_in[1];   // [8,3,9,512,512]
  float* out = (float*)d_out;                 // [8,3,512,512]

  const int grid = KB * KC * (KH / TY);       // 1536 blocks
  dynfilt3x3_kernel<<<grid, 256, 0, stream>>>(img, ker, out);
}